// SpatioTemporalRegressor_55448027791412
// MI455X (gfx1250) — compile-verified
//
#include <hip/hip_runtime.h>
#include <math.h>

#define T_DIM 1000
#define K_DIM 500
#define D_DEPTH 20

typedef __attribute__((ext_vector_type(2))) float v2f;
typedef __attribute__((ext_vector_type(8))) float v8f;

__device__ __forceinline__ float softplus_f(float x) {
    // numerically stable: max(x,0) + log1p(exp(-|x|))
    return fmaxf(x, 0.0f) + log1pf(expf(-fabsf(x)));
}

// ---------------- Phase 1: small prep kernels ----------------

// base[k] = mean_t speeds[t,k] / 10 + base_param[k]
__global__ void base_kernel(const float* __restrict__ speeds,
                            const float* __restrict__ base_param,
                            float* __restrict__ base) {
    int k = blockIdx.x * blockDim.x + threadIdx.x;
    if (k >= K_DIM) return;
    float s = 0.0f;
    for (int t = 0; t < T_DIM; ++t) s += speeds[t * K_DIM + k];
    base[k] = (s / (float)T_DIM) * 0.1f + base_param[k];
}

// pred0.T layout: outP0[k*T + t] = base[k]
__global__ void pred0_kernel(const float* __restrict__ base,
                             float* __restrict__ outP0) {
    int idx = blockIdx.x * blockDim.x + threadIdx.x;
    if (idx >= K_DIM * T_DIM) return;
    __builtin_nontemporal_store(base[idx / T_DIM], &outP0[idx]);
}

// Aeff = Alpha * gsupp
__global__ void aeff_kernel(const float* __restrict__ Alpha,
                            const int* __restrict__ gsupp,
                            float* __restrict__ aeff) {
    int idx = blockIdx.x * blockDim.x + threadIdx.x;
    if (idx >= K_DIM * K_DIM) return;
    aeff[idx] = Alpha[idx] * (float)gsupp[idx];
}

// y[t,j] = sum_{delta=1..min(d,t)} speeds[t-delta, j] * Beta[j] * exp(-delta*Beta[j])
__global__ void y_kernel(const float* __restrict__ speeds,
                         const float* __restrict__ Beta,
                         float* __restrict__ yb) {
    int idx = blockIdx.x * blockDim.x + threadIdx.x;
    if (idx >= T_DIM * K_DIM) return;
    int t = idx / K_DIM;
    int j = idx - t * K_DIM;
    float b = Beta[j];
    int dmax = (t < D_DEPTH) ? t : D_DEPTH;
    float acc = 0.0f;
    for (int delta = 1; delta <= dmax; ++delta) {
        float coef = b * expf(-(float)delta * b);
        acc = fmaf(coef, speeds[(t - delta) * K_DIM + j], acc);
    }
    yb[idx] = acc;
}

// ---------------- Phase 2: dominant streaming kernel ----------------
// raw[t,i] = sum_j Aeff[i,j] * dgraph[t,i,j] * y[t,j]
// Row-sum over j done on the matrix unit: D = A(16x4 of products) x B(4x16 ones) + C.
// dgraph (1 GB, zero reuse) is loaded non-temporally so Aeff (1 MB) and y (2 MB),
// which are re-read ~1000x, stay resident in WGP$/L2.

__device__ __forceinline__ v8f wmma_chunk(const float* __restrict__ dgrow,
                                          const float* __restrict__ aerow,
                                          const float* __restrict__ yrow,
                                          int jj, float mask, v2f bones, v8f acc) {
    v2f dg = __builtin_nontemporal_load((const v2f*)(dgrow + jj)); // streamed once
    v2f ae = *(const v2f*)(aerow + jj);                            // L2-resident
    v2f yv = *(const v2f*)(yrow + jj);                             // L2-resident
    v2f a;
    a.x = dg.x * ae.x * yv.x * mask;
    a.y = dg.y * ae.y * yv.y * mask;
    // 8 args: (neg_a, A, neg_b, B, c_mod, C, reuse_a, reuse_b)
    return __builtin_amdgcn_wmma_f32_16x16x4_f32(false, a, false, bones,
                                                 (short)0, acc, false, false);
}

__global__ __launch_bounds__(128)
void raw_kernel(const float* __restrict__ dgraph,
                const float* __restrict__ aeff,
                const float* __restrict__ yb,
                float* __restrict__ outP1) {
    const int BLOCKS_PER_T = 8;            // 32 tiles / 4 waves per block
    int t    = blockIdx.x / BLOCKS_PER_T;
    int wave = threadIdx.x >> 5;
    int tile = (blockIdx.x % BLOCKS_PER_T) * 4 + wave;   // 0..31
    int lane = threadIdx.x & 31;
    int sub  = lane & 15;                  // row within tile (M)
    int hi   = lane >> 4;                  // 0 -> K=0,1 ; 1 -> K=2,3

    int i0 = tile * 16;
    int i  = i0 + sub;
    float mask = (i < K_DIM) ? 1.0f : 0.0f;
    int ic = (i < K_DIM) ? i : (K_DIM - 1);   // clamp, contribution masked to 0

    const float* dgrow = dgraph + (size_t)t * (K_DIM * K_DIM) + (size_t)ic * K_DIM;
    const float* aerow = aeff + (size_t)ic * K_DIM;
    const float* yrow  = yb + (size_t)t * K_DIM;

    v8f acc0 = {};
    v8f acc1 = {};
    v2f bones; bones.x = 1.0f; bones.y = 1.0f;

    int jo = 2 * hi;
    // 125 chunks of 4 j's: 62 pairs + 1 tail; two accumulators break the RAW chain
    for (int jc = 0; jc < 496; jc += 8) {
        acc0 = wmma_chunk(dgrow, aerow, yrow, jc + jo,     mask, bones, acc0);
        acc1 = wmma_chunk(dgrow, aerow, yrow, jc + 4 + jo, mask, bones, acc1);
    }
    acc0 = wmma_chunk(dgrow, aerow, yrow, 496 + jo, mask, bones, acc0);

    v8f accv = acc0 + acc1;

    // D layout: lane 0 holds M=0..7 (VGPR 0..7) at N=0; lane 16 holds M=8..15 at N=0
    if (sub == 0) {
        int ibase = i0 + hi * 8;
        #pragma unroll
        for (int r = 0; r < 8; ++r) {
            int ii = ibase + r;
            if (ii < K_DIM) {
                float v = (t == 0) ? 0.0f : softplus_f(accv[r]);
                __builtin_nontemporal_store(v, &outP1[(size_t)ii * T_DIM + t]);
            }
        }
    }
}

// ---------------- Phase 3: deterministic loss reduction ----------------

__global__ void loss_partial_kernel(const float* __restrict__ speeds,
                                    const float* __restrict__ outP1,
                                    const float* __restrict__ base,
                                    float* __restrict__ partial) {
    __shared__ float sdata[256];
    float acc = 0.0f;
    int n = T_DIM * K_DIM;
    for (int idx = blockIdx.x * blockDim.x + threadIdx.x; idx < n;
         idx += gridDim.x * blockDim.x) {
        int t = idx / K_DIM;
        int k = idx - t * K_DIM;
        float diff = speeds[idx] - outP1[(size_t)k * T_DIM + t] - base[k];
        acc = fmaf(diff, diff, acc);
    }
    sdata[threadIdx.x] = acc;
    __syncthreads();
    for (int s = 128; s > 0; s >>= 1) {
        if (threadIdx.x < (unsigned)s) sdata[threadIdx.x] += sdata[threadIdx.x + s];
        __syncthreads();
    }
    if (threadIdx.x == 0) partial[blockIdx.x] = sdata[0];
}

__global__ void loss_final_kernel(const float* __restrict__ partial, int n,
                                  float* __restrict__ out) {
    __shared__ float sdata[256];
    float acc = 0.0f;
    for (int i = threadIdx.x; i < n; i += 256) acc += partial[i];
    sdata[threadIdx.x] = acc;
    __syncthreads();
    for (int s = 128; s > 0; s >>= 1) {
        if (threadIdx.x < (unsigned)s) sdata[threadIdx.x] += sdata[threadIdx.x + s];
        __syncthreads();
    }
    if (threadIdx.x == 0) out[0] = sdata[0] / (float)(T_DIM * K_DIM);
}

// ---------------- Launch ----------------

extern "C" void kernel_launch(void* const* d_in, const int* in_sizes, int n_in,
                              void* d_out, int out_size, void* d_ws, size_t ws_size,
                              hipStream_t stream) {
    const float* speeds     = (const float*)d_in[0];
    const float* dgraph     = (const float*)d_in[1];
    const int*   gsupp      = (const int*)d_in[2];
    const float* base_param = (const float*)d_in[3];
    const float* Beta       = (const float*)d_in[4];
    const float* Alpha      = (const float*)d_in[5];
    // d_in[6] is the python scalar d (== D_DEPTH), hardcoded.

    float* ws      = (float*)d_ws;
    float* base    = ws;                         // 512 floats
    float* aeff    = ws + 512;                   // 250112 floats (padded)
    float* yb      = ws + 512 + 250112;          // 500224 floats (padded)
    float* partial = ws + 512 + 250112 + 500224; // 512 floats

    float* out    = (float*)d_out;
    float* outP0  = out + 1;                     // pred0.T [K,T]
    float* outP1  = out + 1 + K_DIM * T_DIM;     // pred1.T [K,T]

    base_kernel<<<(K_DIM + 255) / 256, 256, 0, stream>>>(speeds, base_param, base);
    aeff_kernel<<<(K_DIM * K_DIM + 255) / 256, 256, 0, stream>>>(Alpha, gsupp, aeff);
    y_kernel<<<(T_DIM * K_DIM + 255) / 256, 256, 0, stream>>>(speeds, Beta, yb);
    pred0_kernel<<<(K_DIM * T_DIM + 255) / 256, 256, 0, stream>>>(base, outP0);

    raw_kernel<<<T_DIM * 8, 128, 0, stream>>>(dgraph, aeff, yb, outP1);

    loss_partial_kernel<<<512, 256, 0, stream>>>(speeds, outP1, base, partial);
    loss_final_kernel<<<1, 256, 0, stream>>>(partial, 512, out);
}